// GCN2_16226386444396
// MI455X (gfx1250) — compile-verified
//
#include <hip/hip_runtime.h>

typedef __attribute__((ext_vector_type(2))) float v2f;
typedef __attribute__((ext_vector_type(8))) float v8f;

#define TROW 66  // padded LDS row stride (floats) to avoid 64-bank conflicts

__device__ __forceinline__ v8f wmma_f32(v2f a, v2f b, v8f c) {
  // D = A(16x4 f32) * B(4x16 f32) + C(16x16 f32)
  return __builtin_amdgcn_wmma_f32_16x16x4_f32(false, a, false, b, (short)0, c,
                                               false, false);
}

// Stage a 64x64 fp32 weight matrix into LDS with K-pair interleaving:
// s[(k>>1)*128 + c*2 + (k&1)] = g[k*64 + c]
// so that the WMMA B-fragment (W[k][c], W[k+1][c]) is one contiguous ds_load_b64.
__device__ __forceinline__ void stage_weight64(const float* __restrict__ g,
                                               float* __restrict__ s, int tid) {
  for (int base = tid * 4; base < 64 * 64; base += 128 * 4) {
    float4 v = *(const float4*)(g + base);
    int k = base >> 6;
    int c = base & 63;
    int o = (k >> 1) * 128 + (k & 1);
    s[o + (c + 0) * 2] = v.x;
    s[o + (c + 1) * 2] = v.y;
    s[o + (c + 2) * 2] = v.z;
    s[o + (c + 3) * 2] = v.w;
  }
}

// -------- init: buf0 = x ; stats[0..127] = 0 --------
__global__ void init_copy(const float* __restrict__ x, float* __restrict__ buf0,
                          float* __restrict__ stats, long long total4) {
  long long i = blockIdx.x * (long long)blockDim.x + threadIdx.x;
  if (i < total4) ((float4*)buf0)[i] = ((const float4*)x)[i];
  if (i < 128) stats[i] = 0.f;
}

// -------- edge scatter: agg[dst] += feat[src], 16 threads/edge, float4 each --------
__global__ void scatter_add(const float* __restrict__ feat,
                            const int* __restrict__ src,
                            const int* __restrict__ dst,
                            float* __restrict__ agg, long long e) {
  long long tid = blockIdx.x * (long long)blockDim.x + threadIdx.x;
  if (tid >= e * 16) return;
  long long edge = tid >> 4;
  int part = (int)(tid & 15);
  int s = src[edge];
  int d = dst[edge];
  float4 v = *(const float4*)(feat + (long long)s * 64 + part * 4);
  float* o = agg + (long long)d * 64 + part * 4;
  unsafeAtomicAdd(o + 0, v.x);
  unsafeAtomicAdd(o + 1, v.y);
  unsafeAtomicAdd(o + 2, v.z);
  unsafeAtomicAdd(o + 3, v.w);
}

// -------- MLP1 (WMMA) + fused batchnorm statistics --------
// block = 128 threads (4 waves); wave w handles rows [blk*64 + w*16, +16)
__global__ void __launch_bounds__(128)
gin_mlp1(const float* __restrict__ hin, const float* __restrict__ wa,
         const float* __restrict__ ba, const float* __restrict__ wb,
         const float* __restrict__ bb, float* __restrict__ hout,
         float* __restrict__ gstats, int n) {
  __shared__ float sWa[64 * 64];
  __shared__ float sWb[64 * 64];
  __shared__ float sT[4 * 16 * TROW];
  __shared__ float sStats[128];

  const int tid = threadIdx.x;
  stage_weight64(wa, sWa, tid);
  stage_weight64(wb, sWb, tid);
  sStats[tid] = 0.f;
  __syncthreads();

  const int lane = tid & 31;
  const int wv = tid >> 5;
  const int lh = lane & 15;
  const int hh2 = (lane >> 4) * 2;  // 0 or 2 (K offset for this half-wave)
  const int rowBase = blockIdx.x * 64 + wv * 16;
  int r = rowBase + lh;
  if (r >= n) r = n - 1;  // clamp loads; stores guarded
  const float* hrow = hin + (long long)r * 64;
  float* tt = sT + wv * 16 * TROW;

  // ---- stage 1: t = relu(hin @ Wa + ba), 16x64 tile per wave ----
  v8f acc[4];
#pragma unroll
  for (int ct = 0; ct < 4; ++ct) {
    float bv = ba[ct * 16 + lh];
    v8f a;
#pragma unroll
    for (int j = 0; j < 8; ++j) a[j] = bv;
    acc[ct] = a;
  }
#pragma unroll
  for (int k = 0; k < 64; k += 4) {
    v2f af = *(const v2f*)(hrow + k + hh2);
    const float* brow = sWa + ((k + hh2) >> 1) * 128;
#pragma unroll
    for (int ct = 0; ct < 4; ++ct) {
      v2f bf = *(const v2f*)(brow + (ct * 16 + lh) * 2);
      acc[ct] = wmma_f32(af, bf, acc[ct]);
    }
  }
  // ReLU, bounce C-layout -> A-layout through wave-private LDS tile
#pragma unroll
  for (int ct = 0; ct < 4; ++ct) {
#pragma unroll
    for (int j = 0; j < 8; ++j) {
      float v = acc[ct][j];
      v = v > 0.f ? v : 0.f;
      tt[(j + (hh2 << 2)) * TROW + ct * 16 + lh] = v;
    }
  }

  // ---- stage 2: h1 = t @ Wb + bb ----
  v8f acc2[4];
#pragma unroll
  for (int ct = 0; ct < 4; ++ct) {
    float bv = bb[ct * 16 + lh];
    v8f a;
#pragma unroll
    for (int j = 0; j < 8; ++j) a[j] = bv;
    acc2[ct] = a;
  }
#pragma unroll
  for (int k = 0; k < 64; k += 4) {
    v2f af = *(const v2f*)(tt + lh * TROW + k + hh2);
    const float* brow = sWb + ((k + hh2) >> 1) * 128;
#pragma unroll
    for (int ct = 0; ct < 4; ++ct) {
      v2f bf = *(const v2f*)(brow + (ct * 16 + lh) * 2);
      acc2[ct] = wmma_f32(af, bf, acc2[ct]);
    }
  }

  // ---- epilogue: store h1 + accumulate BN sum / sumsq ----
#pragma unroll
  for (int ct = 0; ct < 4; ++ct) {
    const int col = ct * 16 + lh;
    float s0 = 0.f, q0 = 0.f;
#pragma unroll
    for (int j = 0; j < 8; ++j) {
      int row = rowBase + j + (hh2 << 2);
      float v = acc2[ct][j];
      if (row < n) {
        hout[(long long)row * 64 + col] = v;
        s0 += v;
        q0 += v * v;
      }
    }
    unsafeAtomicAdd(&sStats[col], s0);
    unsafeAtomicAdd(&sStats[64 + col], q0);
  }
  __syncthreads();
  unsafeAtomicAdd(&gstats[tid], sStats[tid]);
}

// -------- BN finalize: stats[128+c]=scale, stats[192+c]=shift --------
__global__ void bn_finalize(const float* __restrict__ gamma,
                            const float* __restrict__ beta,
                            float* __restrict__ stats, float inv_n) {
  int c = threadIdx.x;
  float mean = stats[c] * inv_n;
  float var = stats[64 + c] * inv_n - mean * mean;
  float sc = gamma[c] * rsqrtf(var + 1e-5f);
  stats[128 + c] = sc;
  stats[192 + c] = beta[c] - mean * sc;
}

// -------- BN apply in place, and copy result into agg buffer (layer-2 self term) --------
__global__ void bn_apply(float* __restrict__ h, float* __restrict__ agg,
                         const float* __restrict__ stats, long long total4) {
  long long i = blockIdx.x * (long long)blockDim.x + threadIdx.x;
  if (i >= total4) return;
  int cbase = (int)(i & 15) * 4;
  float4 v = ((float4*)h)[i];
  float4 sc = *(const float4*)(stats + 128 + cbase);
  float4 sh = *(const float4*)(stats + 192 + cbase);
  v.x = v.x * sc.x + sh.x;
  v.y = v.y * sc.y + sh.y;
  v.z = v.z * sc.z + sh.z;
  v.w = v.w * sc.w + sh.w;
  ((float4*)h)[i] = v;
  ((float4*)agg)[i] = v;
}

// -------- MLP2 (WMMA) + final 64x10 linear --------
__global__ void __launch_bounds__(128)
gin_mlp2(const float* __restrict__ hin, const float* __restrict__ wa,
         const float* __restrict__ ba, const float* __restrict__ wb,
         const float* __restrict__ bb, const float* __restrict__ lw,
         const float* __restrict__ lb, float* __restrict__ out, int n) {
  __shared__ float sWa[64 * 64];
  __shared__ float sWb[64 * 64];
  __shared__ float sT[4 * 16 * TROW];
  __shared__ float sLin[64 * 16];  // lin_w zero-padded to 16 cols, pair-interleaved
  __shared__ float sLb[16];

  const int tid = threadIdx.x;
  stage_weight64(wa, sWa, tid);
  stage_weight64(wb, sWb, tid);
  for (int i = tid; i < 64 * 16; i += 128) {
    int k = i >> 4, c = i & 15;
    float v = (c < 10) ? lw[k * 10 + c] : 0.f;
    sLin[(k >> 1) * 32 + c * 2 + (k & 1)] = v;
  }
  if (tid < 16) sLb[tid] = (tid < 10) ? lb[tid] : 0.f;
  __syncthreads();

  const int lane = tid & 31;
  const int wv = tid >> 5;
  const int lh = lane & 15;
  const int hh2 = (lane >> 4) * 2;
  const int rowBase = blockIdx.x * 64 + wv * 16;
  int r = rowBase + lh;
  if (r >= n) r = n - 1;
  const float* hrow = hin + (long long)r * 64;
  float* tt = sT + wv * 16 * TROW;

  // ---- t2 = relu(hin @ Wa + ba) ----
  v8f acc[4];
#pragma unroll
  for (int ct = 0; ct < 4; ++ct) {
    float bv = ba[ct * 16 + lh];
    v8f a;
#pragma unroll
    for (int j = 0; j < 8; ++j) a[j] = bv;
    acc[ct] = a;
  }
#pragma unroll
  for (int k = 0; k < 64; k += 4) {
    v2f af = *(const v2f*)(hrow + k + hh2);
    const float* brow = sWa + ((k + hh2) >> 1) * 128;
#pragma unroll
    for (int ct = 0; ct < 4; ++ct) {
      v2f bf = *(const v2f*)(brow + (ct * 16 + lh) * 2);
      acc[ct] = wmma_f32(af, bf, acc[ct]);
    }
  }
#pragma unroll
  for (int ct = 0; ct < 4; ++ct) {
#pragma unroll
    for (int j = 0; j < 8; ++j) {
      float v = acc[ct][j];
      v = v > 0.f ? v : 0.f;
      tt[(j + (hh2 << 2)) * TROW + ct * 16 + lh] = v;
    }
  }

  // ---- h2 = t2 @ Wb + bb ----
  v8f acc2[4];
#pragma unroll
  for (int ct = 0; ct < 4; ++ct) {
    float bv = bb[ct * 16 + lh];
    v8f a;
#pragma unroll
    for (int j = 0; j < 8; ++j) a[j] = bv;
    acc2[ct] = a;
  }
#pragma unroll
  for (int k = 0; k < 64; k += 4) {
    v2f af = *(const v2f*)(tt + lh * TROW + k + hh2);
    const float* brow = sWb + ((k + hh2) >> 1) * 128;
#pragma unroll
    for (int ct = 0; ct < 4; ++ct) {
      v2f bf = *(const v2f*)(brow + (ct * 16 + lh) * 2);
      acc2[ct] = wmma_f32(af, bf, acc2[ct]);
    }
  }

  // reuse tt for h2 (same-wave LDS ops are in-order: stage-2 reads precede these writes)
#pragma unroll
  for (int ct = 0; ct < 4; ++ct) {
#pragma unroll
    for (int j = 0; j < 8; ++j) {
      tt[(j + (hh2 << 2)) * TROW + ct * 16 + lh] = acc2[ct][j];
    }
  }

  // ---- out = h2 @ lin_w + lin_b (single 16-col tile, cols 10..15 padded) ----
  v8f acc3;
  {
    float bv = sLb[lh];
#pragma unroll
    for (int j = 0; j < 8; ++j) acc3[j] = bv;
  }
#pragma unroll
  for (int k = 0; k < 64; k += 4) {
    v2f af = *(const v2f*)(tt + lh * TROW + k + hh2);
    v2f bf = *(const v2f*)(sLin + ((k + hh2) >> 1) * 32 + lh * 2);
    acc3 = wmma_f32(af, bf, acc3);
  }
  if (lh < 10) {
#pragma unroll
    for (int j = 0; j < 8; ++j) {
      int row = rowBase + j + (hh2 << 2);
      if (row < n) out[(long long)row * 10 + lh] = acc3[j];
    }
  }
}

extern "C" void kernel_launch(void* const* d_in, const int* in_sizes, int n_in,
                              void* d_out, int out_size, void* d_ws,
                              size_t ws_size, hipStream_t stream) {
  const float* x   = (const float*)d_in[0];
  const int*   ei  = (const int*)d_in[1];
  const float* w1a = (const float*)d_in[2];
  const float* b1a = (const float*)d_in[3];
  const float* w1b = (const float*)d_in[4];
  const float* b1b = (const float*)d_in[5];
  const float* gam = (const float*)d_in[6];
  const float* bet = (const float*)d_in[7];
  const float* w2a = (const float*)d_in[8];
  const float* b2a = (const float*)d_in[9];
  const float* w2b = (const float*)d_in[10];
  const float* b2b = (const float*)d_in[11];
  const float* lw  = (const float*)d_in[12];
  const float* lb  = (const float*)d_in[13];
  float* out = (float*)d_out;

  const int n = in_sizes[0] / 64;
  const long long e = in_sizes[1] / 2;
  const int* src = ei;
  const int* dst = ei + e;

  float* buf0  = (float*)d_ws;                 // x+agg / h1bn+agg2   [n*64]
  float* buf1  = buf0 + (long long)n * 64;     // h1 / h1bn           [n*64]
  float* stats = buf1 + (long long)n * 64;     // sum,sumsq,scale,shift [256]

  const long long total4 = (long long)n * 16;
  const int initBlocks = (int)((total4 + 255) / 256);
  const long long stotal = e * 16;
  const int sBlocks = (int)((stotal + 255) / 256);
  const int mBlocks = (n + 63) / 64;

  // Layer 1
  init_copy<<<initBlocks, 256, 0, stream>>>(x, buf0, stats, total4);
  scatter_add<<<sBlocks, 256, 0, stream>>>(x, src, dst, buf0, e);
  gin_mlp1<<<mBlocks, 128, 0, stream>>>(buf0, w1a, b1a, w1b, b1b, buf1, stats, n);
  bn_finalize<<<1, 64, 0, stream>>>(gam, bet, stats, 1.0f / (float)n);
  bn_apply<<<initBlocks, 256, 0, stream>>>(buf1, buf0, stats, total4);
  // Layer 2
  scatter_add<<<sBlocks, 256, 0, stream>>>(buf1, src, dst, buf0, e);
  gin_mlp2<<<mBlocks, 128, 0, stream>>>(buf0, w2a, b2a, w2b, b2b, lw, lb, out, n);
}